// TransformerConv_74749610819921
// MI455X (gfx1250) — compile-verified
//
#include <hip/hip_runtime.h>
#include <math.h>

// ---------------------------------------------------------------------------
// TransformerConv (edge-conditioned graph attention), B=4 N=256 F=128 E=4 H=4 D=32
//
// Algebraic restructuring (avoids the [B,N,N,H,D] = 134MB 'e' tensor):
//   qk[b,i,h]    = <q_bih, k_bih>
//   qWe[b,i,h,e] = <q_bih, We[e, h*D:(h+1)*D]>
//   logit        = (A*qk + sum_e adj_e * qWe_e)/sqrt(D) - 1e10*(1-A)
//   alpha        = softmax_j(logit)
//   aw[b,i,h,e]  = sum_j alpha * adj_e
//   out          = (alpha*A) @ V_h            (WMMA GEMM, per (b,h): 256x256 x 256x32)
//                + aw @ We_h                  (rank-4 correction)
//                + x@Wroot + broot            (WMMA GEMM)
//
// All matrix work uses V_WMMA_F32_16X16X4_F32 (numerically-exact fp32 WMMA).
// Problem is memory/latency bound (~20MB traffic @ 23.3TB/s ~ 1us), so fp32
// matrix cores are the right choice: exact, and nowhere near the bottleneck.
// The aw-reduction uses wave32 shuffle pre-reduction (no same-address LDS
// atomics -> no serialization).
// ---------------------------------------------------------------------------

#define Bn 4
#define Nn 256
#define FIN 128
#define En 4
#define Hn 4
#define Dn 32
#define HD 128          // H*D
#define BN (Bn*Nn)      // 1024

// workspace layout (float offsets); total = 1,593,344 floats = ~6.4 MB
#define OFF_Q    0
#define OFF_K    (OFF_Q    + BN*HD)      // 131072
#define OFF_V    (OFF_K    + BN*HD)      // 262144
#define OFF_BASE (OFF_V    + BN*HD)      // 393216  (x@Wroot+broot, then += edge term)
#define OFF_QK   (OFF_BASE + BN*HD)      // 524288  [BN][H]
#define OFF_QWE  (OFF_QK   + BN*Hn)      // 528384  [BN][H][E]
#define OFF_AA   (OFF_QWE  + BN*Hn*En)   // 544768  [B][H][N][N] = alpha*A, GEMM A-matrix

typedef float v2f __attribute__((ext_vector_type(2)));
typedef float v8f __attribute__((ext_vector_type(8)));

static __device__ inline v8f wmma_f32_4(v2f a, v2f b, v8f c) {
    // D = A(16x4) * B(4x16) + C, fp32 matrix core
    return __builtin_amdgcn_wmma_f32_16x16x4_f32(
        /*neg_a=*/false, a, /*neg_b=*/false, b,
        /*c_mod=*/(short)0, c, /*reuse_a=*/false, /*reuse_b=*/false);
}

// ---------------------------------------------------------------------------
// Kernel 1: four projection GEMMs  Y = x @ W + bias
//   x:[1024,128] W:[128,128] -> q / k / v / base   (base = root projection)
// One wave per 16x16 output tile; 4 mats * 64 * 8 tiles = 2048 waves = 256 blocks.
// ---------------------------------------------------------------------------
__global__ __launch_bounds__(256) void proj_kernel(
    const float* __restrict__ x,
    const float* __restrict__ Wq, const float* __restrict__ bq,
    const float* __restrict__ Wk, const float* __restrict__ bk,
    const float* __restrict__ Wv, const float* __restrict__ bv,
    const float* __restrict__ Wr, const float* __restrict__ br,
    float* __restrict__ ws)
{
    const int wave = blockIdx.x * 8 + (threadIdx.x >> 5);
    const int lane = threadIdx.x & 31;
    const int mat  = wave >> 9;          // 512 tiles per matrix
    const int tile = wave & 511;
    const int m0   = (tile >> 3) * 16;   // row tile (0..63)
    const int n0   = (tile & 7) * 16;    // col tile (0..7)

    const float* W;  const float* bias;  float* out;
    switch (mat) {
        case 0:  W = Wq; bias = bq; out = ws + OFF_Q;    break;
        case 1:  W = Wk; bias = bk; out = ws + OFF_K;    break;
        case 2:  W = Wv; bias = bv; out = ws + OFF_V;    break;
        default: W = Wr; bias = br; out = ws + OFF_BASE; break;
    }

    const int c  = lane & 15;            // A-row / B-col / CD-col
    const int kh = lane >> 4;            // K half-select

    v8f acc = {0.f,0.f,0.f,0.f,0.f,0.f,0.f,0.f};
    for (int k0 = 0; k0 < FIN; k0 += 4) {
        v2f a, b;
        const float* ap = x + (m0 + c) * FIN + k0 + 2 * kh;   // A[r][k0+2kh +0/+1]
        a[0] = ap[0];
        a[1] = ap[1];
        const float* bp = W + (k0 + 2 * kh) * HD + n0 + c;    // B[k0+2kh +0/+1][col]
        b[0] = bp[0];
        b[1] = bp[HD];
        acc = wmma_f32_4(a, b, acc);
    }
    const float bb = bias[n0 + c];
    #pragma unroll
    for (int g = 0; g < 8; ++g)
        out[(m0 + g + 8 * kh) * HD + n0 + c] = acc[g] + bb;
}

// ---------------------------------------------------------------------------
// Kernel 2: qk[b,i,h] = <q,k>  and  qWe[b,i,h,e] = <q, We[e,hD:]>
// One block per (b,i); one wave per head; lane = d. Wave32 shuffle reductions.
// ---------------------------------------------------------------------------
__global__ __launch_bounds__(128) void qk_qwe_kernel(
    const float* __restrict__ q, const float* __restrict__ k,
    const float* __restrict__ We,
    float* __restrict__ qk, float* __restrict__ qWe)
{
    const int bi = blockIdx.x;             // b*N + i
    const int h  = threadIdx.x >> 5;
    const int d  = threadIdx.x & 31;

    const float qv = q[bi * HD + h * Dn + d];
    const float kv = k[bi * HD + h * Dn + d];

    float s = qv * kv;
    #pragma unroll
    for (int m = 16; m; m >>= 1) s += __shfl_xor(s, m, 32);
    if (d == 0) qk[bi * Hn + h] = s;

    #pragma unroll
    for (int e = 0; e < En; ++e) {
        float t = qv * We[e * HD + h * Dn + d];
        #pragma unroll
        for (int m = 16; m; m >>= 1) t += __shfl_xor(t, m, 32);
        if (d == 0) qWe[(bi * Hn + h) * En + e] = t;
    }
}

// ---------------------------------------------------------------------------
// Kernel 3: per (b,i): logits for all j,h, softmax over j, write alpha*A
// (transposed into [B,H,N,N] for the GEMM), wave-shuffle-reduce aw[h][e],
// and fold the aw@We_h edge correction into base[b,i,:].
// One block (256 threads = 8 waves) per (b,i); thread = source node j.
// ---------------------------------------------------------------------------
__global__ __launch_bounds__(256) void attn_kernel(
    const float* __restrict__ adj, const float* __restrict__ We,
    const float* __restrict__ qk,  const float* __restrict__ qWe,
    float* __restrict__ alphaA, float* __restrict__ base)
{
    const int bi   = blockIdx.x;          // b*N + i
    const int b    = bi >> 8;
    const int i    = bi & 255;
    const int j    = threadIdx.x;
    const int lane = j & 31;
    const int wid  = j >> 5;

    __shared__ float redmax[Hn][8];
    __shared__ float redsum[Hn][8];
    __shared__ float s_aw[Hn][En][8];    // per-wave partials of sum_j alpha*adj_e

    // adj[b,i,j,0..3] in one 16B load
    const float4 aj = ((const float4*)adj)[bi * Nn + j];
    const float ssum = aj.x + aj.y + aj.z + aj.w;
    const float A = (fabsf(ssum) > 0.0f) ? 1.0f : 0.0f;
    const float inv = 0.17677669529663687f;   // 1/sqrt(D=32)

    float logit[Hn];
    #pragma unroll
    for (int h = 0; h < Hn; ++h) {
        const float* qw = &qWe[(bi * Hn + h) * En];
        float l = A * qk[bi * Hn + h]
                + aj.x * qw[0] + aj.y * qw[1] + aj.z * qw[2] + aj.w * qw[3];
        logit[h] = l * inv + (-1e10f) * (1.0f - A);
    }

    // max over j (wave reduce, then 8 partials in LDS)
    #pragma unroll
    for (int h = 0; h < Hn; ++h) {
        float m = logit[h];
        #pragma unroll
        for (int msk = 16; msk; msk >>= 1) m = fmaxf(m, __shfl_xor(m, msk, 32));
        if (lane == 0) redmax[h][wid] = m;
    }
    __syncthreads();

    float p[Hn];
    #pragma unroll
    for (int h = 0; h < Hn; ++h) {
        float m = redmax[h][0];
        #pragma unroll
        for (int w = 1; w < 8; ++w) m = fmaxf(m, redmax[h][w]);
        p[h] = __expf(logit[h] - m);
        float s = p[h];
        #pragma unroll
        for (int msk = 16; msk; msk >>= 1) s += __shfl_xor(s, msk, 32);
        if (lane == 0) redsum[h][wid] = s;
    }
    __syncthreads();

    #pragma unroll
    for (int h = 0; h < Hn; ++h) {
        float s = 0.f;
        #pragma unroll
        for (int w = 0; w < 8; ++w) s += redsum[h][w];
        const float alpha = p[h] / s;
        // GEMM A-matrix, row-major [B,H,N(i),N(j)]
        alphaA[((b * Hn + h) * Nn + i) * Nn + j] = alpha * A;

        // aw[h][e] partials = sum_j alpha * adj_e  (alpha, NOT alpha*A:
        // the e-bias contribution is unmasked in the reference).
        // Wave32 shuffle pre-reduction: no same-address LDS atomics.
        float c0 = alpha * aj.x, c1 = alpha * aj.y;
        float c2 = alpha * aj.z, c3 = alpha * aj.w;
        #pragma unroll
        for (int msk = 16; msk; msk >>= 1) {
            c0 += __shfl_xor(c0, msk, 32);
            c1 += __shfl_xor(c1, msk, 32);
            c2 += __shfl_xor(c2, msk, 32);
            c3 += __shfl_xor(c3, msk, 32);
        }
        if (lane == 0) {
            s_aw[h][0][wid] = c0;
            s_aw[h][1][wid] = c1;
            s_aw[h][2][wid] = c2;
            s_aw[h][3][wid] = c3;
        }
    }
    __syncthreads();

    // base[b,i,h*D+d] += sum_e aw[h][e] * We[e, h*D+d]
    if (threadIdx.x < HD) {
        const int h = threadIdx.x >> 5;
        const int d = threadIdx.x & 31;
        float aw[En];
        #pragma unroll
        for (int e = 0; e < En; ++e) {
            float s = 0.f;
            #pragma unroll
            for (int w = 0; w < 8; ++w) s += s_aw[h][e][w];
            aw[e] = s;
        }
        const float ecorr = aw[0] * We[0 * HD + h * Dn + d]
                          + aw[1] * We[1 * HD + h * Dn + d]
                          + aw[2] * We[2 * HD + h * Dn + d]
                          + aw[3] * We[3 * HD + h * Dn + d];
        base[bi * HD + h * Dn + d] += ecorr;
    }
}

// ---------------------------------------------------------------------------
// Kernel 4: out[b,i,h*D:] = base[b,i,h*D:] + (alphaA[b,h] @ V[b,:,h*D:])
// Per (b,h): [256x256]x[256x32]. One wave per 16x16 tile: 4*4*16*2 = 512
// waves = 64 blocks. Accumulator initialized from base (C operand).
// ---------------------------------------------------------------------------
__global__ __launch_bounds__(256) void aggregate_kernel(
    const float* __restrict__ alphaA, const float* __restrict__ v,
    const float* __restrict__ base, float* __restrict__ out)
{
    const int wave = blockIdx.x * 8 + (threadIdx.x >> 5);
    const int lane = threadIdx.x & 31;
    const int dt = wave & 1;            // which 16-col half of D=32
    const int i0 = ((wave >> 1) & 15) * 16;
    const int h  = (wave >> 5) & 3;
    const int b  = wave >> 7;
    const int c0 = h * Dn + dt * 16;

    const int c  = lane & 15;
    const int kh = lane >> 4;

    const float* AA = alphaA + (b * Hn + h) * Nn * Nn;
    const float* V  = v + b * Nn * HD;

    v8f acc;
    #pragma unroll
    for (int g = 0; g < 8; ++g)
        acc[g] = base[(b * Nn + i0 + g + 8 * kh) * HD + c0 + c];

    for (int k0 = 0; k0 < Nn; k0 += 4) {
        v2f a, bb;
        const float* ap = AA + (i0 + c) * Nn + k0 + 2 * kh;
        a[0] = ap[0];
        a[1] = ap[1];
        const float* bp = V + (k0 + 2 * kh) * HD + c0 + c;
        bb[0] = bp[0];
        bb[1] = bp[HD];
        acc = wmma_f32_4(a, bb, acc);
    }

    #pragma unroll
    for (int g = 0; g < 8; ++g)
        out[(b * Nn + i0 + g + 8 * kh) * HD + c0 + c] = acc[g];
}

// ---------------------------------------------------------------------------
extern "C" void kernel_launch(void* const* d_in, const int* in_sizes, int n_in,
                              void* d_out, int out_size, void* d_ws, size_t ws_size,
                              hipStream_t stream) {
    (void)in_sizes; (void)n_in; (void)out_size; (void)ws_size;
    const float* x     = (const float*)d_in[0];
    const float* adj   = (const float*)d_in[1];
    const float* Wq    = (const float*)d_in[2];
    const float* bq    = (const float*)d_in[3];
    const float* Wk    = (const float*)d_in[4];
    const float* bk    = (const float*)d_in[5];
    const float* Wv    = (const float*)d_in[6];
    const float* bv    = (const float*)d_in[7];
    const float* We    = (const float*)d_in[8];
    const float* Wroot = (const float*)d_in[9];
    const float* broot = (const float*)d_in[10];

    float* ws = (float*)d_ws;   // needs ~6.4 MB; fully rewritten every call

    proj_kernel<<<256, 256, 0, stream>>>(x, Wq, bq, Wk, bk, Wv, bv, Wroot, broot, ws);
    qk_qwe_kernel<<<BN, 128, 0, stream>>>(ws + OFF_Q, ws + OFF_K, We,
                                          ws + OFF_QK, ws + OFF_QWE);
    attn_kernel<<<BN, 256, 0, stream>>>(adj, We, ws + OFF_QK, ws + OFF_QWE,
                                        ws + OFF_AA, ws + OFF_BASE);
    aggregate_kernel<<<64, 256, 0, stream>>>(ws + OFF_AA, ws + OFF_V,
                                             ws + OFF_BASE, (float*)d_out);
}